// FastWeightMemory_2645699855028
// MI455X (gfx1250) — compile-verified
//
#include <hip/hip_runtime.h>

typedef __attribute__((ext_vector_type(2))) float v2f;
typedef __attribute__((ext_vector_type(8))) float v8f;

#define BATCH  256
#define DIM    1024
#define KVDIM  2048

// ---------------------------------------------------------------------------
// Kernel 1: kv = x @ W^T + bias   (256 x 2048, f32, via V_WMMA_F32_16X16X4_F32)
//
// Block = 128 threads = 4 waves. Each wave owns a 16x64 strip of kv:
//   rows  [blockIdx.y*16, +16)
//   cols  [blockIdx.x*256 + wave*64, +64)   as 4 adjacent 16x16 C tiles.
// Per k-step of 4: one A load (shared by 4 tiles) + 4 B loads + 4 WMMAs.
//
// VGPR layouts (wave32, ISA 7.12.2):
//   A 16x4 : lane r=lane&15 -> M=r ; VGPR{0,1} hold K = 2*half + {0,1}
//   B 4x16 : lane r        -> N=r ; VGPR{0,1} hold K = 2*half + {0,1}
//            (B[k][n] = W[col0+n][k], so both A and B are contiguous b64 loads)
//   C 16x16: lane r -> N=r ; VGPR g -> M = g + 8*half
// ---------------------------------------------------------------------------
__global__ __launch_bounds__(128) void kv_gemm_wmma(
    const float* __restrict__ x,     // [BATCH, DIM]
    const float* __restrict__ W,     // [KVDIM, DIM]
    const float* __restrict__ bias,  // [KVDIM]
    float* __restrict__ kv)          // [BATCH, KVDIM]
{
  const int wave = threadIdx.x >> 5;
  const int lane = threadIdx.x & 31;
  const int half = lane >> 4;      // 0: K={0,1}, 1: K={2,3}
  const int r    = lane & 15;

  const int row0 = blockIdx.y * 16;
  const int col0 = blockIdx.x * 256 + wave * 64;

  // per-lane base pointers (two consecutive f32 per load)
  const float* ap  = x + (size_t)(row0 + r) * DIM + 2 * half;
  const float* bp0 = W + (size_t)(col0 +  0 + r) * DIM + 2 * half;
  const float* bp1 = W + (size_t)(col0 + 16 + r) * DIM + 2 * half;
  const float* bp2 = W + (size_t)(col0 + 32 + r) * DIM + 2 * half;
  const float* bp3 = W + (size_t)(col0 + 48 + r) * DIM + 2 * half;

  v8f c0 = {}, c1 = {}, c2 = {}, c3 = {};

#pragma unroll 4
  for (int kk = 0; kk < DIM; kk += 4) {
    v2f a  = *(const v2f*)(ap  + kk);
    v2f b0 = *(const v2f*)(bp0 + kk);
    v2f b1 = *(const v2f*)(bp1 + kk);
    v2f b2 = *(const v2f*)(bp2 + kk);
    v2f b3 = *(const v2f*)(bp3 + kk);
    // 8 args: (neg_a, A, neg_b, B, c_mod, C, reuse_a, reuse_b)
    c0 = __builtin_amdgcn_wmma_f32_16x16x4_f32(false, a, false, b0, (short)0, c0, false, false);
    c1 = __builtin_amdgcn_wmma_f32_16x16x4_f32(false, a, false, b1, (short)0, c1, false, false);
    c2 = __builtin_amdgcn_wmma_f32_16x16x4_f32(false, a, false, b2, (short)0, c2, false, false);
    c3 = __builtin_amdgcn_wmma_f32_16x16x4_f32(false, a, false, b3, (short)0, c3, false, false);
  }

  // Epilogue: bias add (per output column N = tile_col0 + r) and store.
  const float bb0 = bias[col0 +  0 + r];
  const float bb1 = bias[col0 + 16 + r];
  const float bb2 = bias[col0 + 32 + r];
  const float bb3 = bias[col0 + 48 + r];

#pragma unroll
  for (int g = 0; g < 8; ++g) {
    const int m = g + 8 * half;
    float* dst = kv + (size_t)(row0 + m) * KVDIM + col0 + r;
    dst[ 0] = c0[g] + bb0;
    dst[16] = c1[g] + bb1;
    dst[32] = c2[g] + bb2;
    dst[48] = c3[g] + bb3;
  }
}

// ---------------------------------------------------------------------------
// Kernel 2: per batch row b:
//   s      = dot(kv[b, 0:DIM], q[b, :])          (block-wide reduction)
//   out[b] = kv[b, DIM:2*DIM] * s
// One 256-thread block (8 waves) per batch row.
// ---------------------------------------------------------------------------
__global__ __launch_bounds__(256) void attend_scale(
    const float* __restrict__ kv,   // [BATCH, KVDIM]
    const float* __restrict__ q,    // [BATCH, DIM]
    float* __restrict__ out)        // [BATCH, DIM]
{
  __shared__ float red[8];
  __shared__ float s_bcast;

  const int b   = blockIdx.x;
  const int tid = threadIdx.x;
  const float* krow = kv + (size_t)b * KVDIM;
  const float* vrow = krow + DIM;
  const float* qrow = q + (size_t)b * DIM;

  float p = 0.0f;
#pragma unroll
  for (int j = tid; j < DIM; j += 256) p += krow[j] * qrow[j];

  // wave32 tree reduction
#pragma unroll
  for (int off = 16; off > 0; off >>= 1) p += __shfl_down(p, off, 32);

  if ((tid & 31) == 0) red[tid >> 5] = p;
  __syncthreads();
  if (tid == 0) {
    float t = 0.0f;
#pragma unroll
    for (int i = 0; i < 8; ++i) t += red[i];
    s_bcast = t;
  }
  __syncthreads();
  const float s = s_bcast;

#pragma unroll
  for (int i = tid; i < DIM; i += 256) out[(size_t)b * DIM + i] = vrow[i] * s;
}

// ---------------------------------------------------------------------------
extern "C" void kernel_launch(void* const* d_in, const int* in_sizes, int n_in,
                              void* d_out, int out_size, void* d_ws, size_t ws_size,
                              hipStream_t stream) {
  const float* x    = (const float*)d_in[0];  // [256,1024]
  const float* qry  = (const float*)d_in[1];  // [256,1024]
  const float* W    = (const float*)d_in[2];  // [2048,1024]
  const float* bias = (const float*)d_in[3];  // [2048]
  float* out = (float*)d_out;                 // [256,1024]
  float* kv  = (float*)d_ws;                  // [256,2048] = 2 MiB scratch

  dim3 g1(KVDIM / 256, BATCH / 16);           // 8 x 16 blocks, 4 waves each
  kv_gemm_wmma<<<g1, 128, 0, stream>>>(x, W, bias, kv);
  attend_scale<<<BATCH, 256, 0, stream>>>(kv, qry, out);
}